// LSTMDecoder_6244882448967
// MI455X (gfx1250) — compile-verified
//
#include <hip/hip_runtime.h>
#include <hip/hip_bf16.h>
#include <math.h>

// Problem constants (match reference)
#define B_ 32
#define T_ 64
#define D_ 512
#define H_ 512
#define V_ 32000
#define EOS_ 3
#define PAD_ 0

typedef __attribute__((ext_vector_type(2))) float v2f;
typedef __attribute__((ext_vector_type(8))) float v8f;

// ---------------------------------------------------------------------------
// fp32 WMMA wrapper: D(16x16) = A(16x4) * B(4x16) + C   (CDNA5 V_WMMA_F32_16X16X4_F32)
// ---------------------------------------------------------------------------
__device__ __forceinline__ v8f wmma4(v2f a, v2f b, v8f c) {
  // 8 args: (neg_a, A, neg_b, B, c_mod, C, reuse_a, reuse_b)
  return __builtin_amdgcn_wmma_f32_16x16x4_f32(
      false, a, false, b, (short)0, c, false, false);
}

// ---------------------------------------------------------------------------
// Init: copy h/c encode state, x0 = input_seq[:,0,:], clear finished flags.
// ---------------------------------------------------------------------------
__global__ __launch_bounds__(256) void init_kernel(
    const float* __restrict__ input_seq, const float* __restrict__ h_enc,
    const float* __restrict__ c_enc, float* __restrict__ h0,
    float* __restrict__ h1, float* __restrict__ c0, float* __restrict__ c1,
    float* __restrict__ x, int* __restrict__ finished) {
  int i = blockIdx.x * blockDim.x + threadIdx.x;
  const int N = B_ * H_;  // 16384
  if (i < N) {
    h0[i] = h_enc[i];
    h1[i] = h_enc[N + i];
    c0[i] = c_enc[i];
    c1[i] = c_enc[N + i];
    int b = i >> 9, d = i & (D_ - 1);
    x[i] = input_seq[(size_t)b * T_ * D_ + d];  // seq[0]
  }
  if (i < B_) finished[i] = 0;
}

// ---------------------------------------------------------------------------
// One LSTM cell for all 32 batch rows, 16 hidden columns per block.
// g = x@Wih^T + h@Whh^T + bih + bhh ; gates i,f,g,o ; update h,c.
// Block: 256 threads = 8 waves; wave = (mTile in {0,1}) x (gate in {0..3}).
// ---------------------------------------------------------------------------
__global__ __launch_bounds__(256) void lstm_layer_kernel(
    const float* __restrict__ x,     // [32,512] layer input
    const float* __restrict__ h_in,  // [32,512]
    float* __restrict__ c,           // [32,512] in/out
    const float* __restrict__ Wih,   // [2048,512] row-major
    const float* __restrict__ Whh,   // [2048,512]
    const float* __restrict__ bih,   // [2048]
    const float* __restrict__ bhh,   // [2048]
    float* __restrict__ h_out)       // [32,512]
{
  __shared__ float gl[4][32][16];  // gate pre-activations

  const int tid = threadIdx.x;
  const int lane = tid & 31;
  const int wave = tid >> 5;
  const int mTile = wave & 1;   // batch rows 0-15 / 16-31
  const int gate = wave >> 1;   // i,f,g,o
  const int n0 = blockIdx.x * 16;         // hidden-column base
  const int gcol = gate * H_ + n0;        // gate-matrix column base

  const int lm = lane & 15;
  const int ko = (lane >> 4) * 2;         // K sub-offset per ISA A/B layout

  const int arow = mTile * 16 + lm;       // A: batch row
  const int brow = gcol + lm;             // B: weight row (= output column)

  v8f a0 = {}, a1 = {}, a2 = {}, a3 = {};

  // ---- x @ Wih^T ----
  {
    const float* aB = x + arow * D_ + ko;
    const float* bB = Wih + (size_t)brow * D_ + ko;
#pragma unroll 4
    for (int k = 0; k < D_; k += 16) {
      v2f xa0 = *(const v2f*)(aB + k);      v2f wb0 = *(const v2f*)(bB + k);
      v2f xa1 = *(const v2f*)(aB + k + 4);  v2f wb1 = *(const v2f*)(bB + k + 4);
      v2f xa2 = *(const v2f*)(aB + k + 8);  v2f wb2 = *(const v2f*)(bB + k + 8);
      v2f xa3 = *(const v2f*)(aB + k + 12); v2f wb3 = *(const v2f*)(bB + k + 12);
      a0 = wmma4(xa0, wb0, a0);
      a1 = wmma4(xa1, wb1, a1);
      a2 = wmma4(xa2, wb2, a2);
      a3 = wmma4(xa3, wb3, a3);
    }
  }
  // ---- h @ Whh^T ----
  {
    const float* aB = h_in + arow * H_ + ko;
    const float* bB = Whh + (size_t)brow * H_ + ko;
#pragma unroll 4
    for (int k = 0; k < H_; k += 16) {
      v2f xa0 = *(const v2f*)(aB + k);      v2f wb0 = *(const v2f*)(bB + k);
      v2f xa1 = *(const v2f*)(aB + k + 4);  v2f wb1 = *(const v2f*)(bB + k + 4);
      v2f xa2 = *(const v2f*)(aB + k + 8);  v2f wb2 = *(const v2f*)(bB + k + 8);
      v2f xa3 = *(const v2f*)(aB + k + 12); v2f wb3 = *(const v2f*)(bB + k + 12);
      a0 = wmma4(xa0, wb0, a0);
      a1 = wmma4(xa1, wb1, a1);
      a2 = wmma4(xa2, wb2, a2);
      a3 = wmma4(xa3, wb3, a3);
    }
  }
  v8f acc = (a0 + a1) + (a2 + a3);

  // C/D layout: element r -> (M = r + 8*(lane>=16), N = lane&15) within tile
  const int mhi = (lane >= 16) ? 8 : 0;
#pragma unroll
  for (int r = 0; r < 8; ++r) {
    gl[gate][mTile * 16 + r + mhi][lm] = acc[r];
  }
  __syncthreads();

  // Elementwise gate math: 32 rows x 16 cols = 512 entries, 2 per thread
  for (int e = tid; e < 32 * 16; e += 256) {
    int m = e >> 4, j = e & 15;
    int col = n0 + j;
    float gi = gl[0][m][j] + bih[col] + bhh[col];
    float gf = gl[1][m][j] + bih[H_ + col] + bhh[H_ + col];
    float gg = gl[2][m][j] + bih[2 * H_ + col] + bhh[2 * H_ + col];
    float go = gl[3][m][j] + bih[3 * H_ + col] + bhh[3 * H_ + col];
    float is = 1.0f / (1.0f + expf(-gi));
    float fs = 1.0f / (1.0f + expf(-gf));
    float os = 1.0f / (1.0f + expf(-go));
    float cOld = c[m * H_ + col];
    float c2 = fs * cOld + is * tanhf(gg);
    float h2 = os * tanhf(c2);
    c[m * H_ + col] = c2;
    h_out[m * H_ + col] = h2;
  }
}

// ---------------------------------------------------------------------------
// logits[b, n] = h1[b,:] @ fc_w[n,:] + fc_b[n], written to out[b, t, n].
// Block: 64 threads = 2 waves (one per 16-row batch tile); 16 columns/block.
// ---------------------------------------------------------------------------
__global__ __launch_bounds__(64) void logits_kernel(
    const float* __restrict__ h,    // [32,512]
    const float* __restrict__ fcw,  // [32000,512]
    const float* __restrict__ fcb,  // [32000]
    float* __restrict__ out,        // [B,T,V]
    int t) {
  const int tid = threadIdx.x;
  const int lane = tid & 31;
  const int mTile = tid >> 5;  // 0 or 1
  const int n0 = blockIdx.x * 16;

  const int lm = lane & 15;
  const int ko = (lane >> 4) * 2;

  const float* aB = h + (mTile * 16 + lm) * H_ + ko;
  const float* bB = fcw + (size_t)(n0 + lm) * H_ + ko;

  v8f a0 = {}, a1 = {}, a2 = {}, a3 = {};
#pragma unroll 4
  for (int k = 0; k < H_; k += 16) {
    v2f xa0 = *(const v2f*)(aB + k);      v2f wb0 = *(const v2f*)(bB + k);
    v2f xa1 = *(const v2f*)(aB + k + 4);  v2f wb1 = *(const v2f*)(bB + k + 4);
    v2f xa2 = *(const v2f*)(aB + k + 8);  v2f wb2 = *(const v2f*)(bB + k + 8);
    v2f xa3 = *(const v2f*)(aB + k + 12); v2f wb3 = *(const v2f*)(bB + k + 12);
    a0 = wmma4(xa0, wb0, a0);
    a1 = wmma4(xa1, wb1, a1);
    a2 = wmma4(xa2, wb2, a2);
    a3 = wmma4(xa3, wb3, a3);
  }
  v8f acc = (a0 + a1) + (a2 + a3);

  const int N = n0 + lm;
  const float bias = fcb[N];
  const int mhi = (lane >= 16) ? 8 : 0;
#pragma unroll
  for (int r = 0; r < 8; ++r) {
    int b = mTile * 16 + r + mhi;  // batch row
    out[((size_t)b * T_ + t) * V_ + N] = acc[r] + bias;
  }
}

// ---------------------------------------------------------------------------
// JAX threefry2x32 (exact round/key-injection schedule)
// ---------------------------------------------------------------------------
__device__ __forceinline__ unsigned rotl32(unsigned x, int r) {
  return (x << r) | (x >> (32 - r));
}
__device__ __forceinline__ void tf2x32(unsigned k0, unsigned k1, unsigned x0,
                                       unsigned x1, unsigned& o0,
                                       unsigned& o1) {
  unsigned ks2 = k0 ^ k1 ^ 0x1BD11BDAu;
  x0 += k0; x1 += k1;
#define TF_ROUND(r) { x0 += x1; x1 = rotl32(x1, (r)); x1 ^= x0; }
  TF_ROUND(13) TF_ROUND(15) TF_ROUND(26) TF_ROUND(6)   x0 += k1;  x1 += ks2 + 1u;
  TF_ROUND(17) TF_ROUND(29) TF_ROUND(16) TF_ROUND(24)  x0 += ks2; x1 += k0 + 2u;
  TF_ROUND(13) TF_ROUND(15) TF_ROUND(26) TF_ROUND(6)   x0 += k0;  x1 += k1 + 3u;
  TF_ROUND(17) TF_ROUND(29) TF_ROUND(16) TF_ROUND(24)  x0 += k1;  x1 += ks2 + 4u;
  TF_ROUND(13) TF_ROUND(15) TF_ROUND(26) TF_ROUND(6)   x0 += ks2; x1 += k0 + 5u;
#undef TF_ROUND
  o0 = x0; o1 = x1;
}

// ---------------------------------------------------------------------------
// Gumbel-max categorical sample per batch row + PAD/EOS logic + embed feedback
// Block b handles row b; 256 threads scan V=32000 logits.
// ---------------------------------------------------------------------------
__global__ __launch_bounds__(256) void sample_embed_kernel(
    const float* __restrict__ logits_all,  // d_out base [B,T,V]
    const float* __restrict__ embed,       // [V, D]
    float* __restrict__ x_next,            // [32,512]
    int* __restrict__ finished,            // [32]
    float* __restrict__ tokens_out,        // d_out tokens area [B,T]
    int t) {
  const int b = blockIdx.x;
  const int tid = threadIdx.x;

  // folded key = threefry(key(42), (0, t))
  unsigned fk0, fk1;
  tf2x32(0u, 42u, 0u, (unsigned)t, fk0, fk1);

  const float* lrow = logits_all + ((size_t)b * T_ + t) * V_;
  const unsigned half = (unsigned)(B_ * V_) / 2u;  // 512000
  const float tiny = 1.17549435e-38f;

  float bestv = -INFINITY;
  int besti = 0;
  for (int v = tid; v < V_; v += 256) {
    unsigned flat = (unsigned)(b * V_ + v);
    unsigned o0, o1, bits;
    if (flat < half) { tf2x32(fk0, fk1, flat, flat + half, o0, o1); bits = o0; }
    else             { tf2x32(fk0, fk1, flat - half, flat, o0, o1); bits = o1; }
    float u = __uint_as_float((bits >> 9) | 0x3f800000u) - 1.0f;
    u = fmaxf(tiny, u + tiny);
    float g = -logf(-logf(u));
    float keyv = lrow[v] + g;
    if (keyv > bestv) { bestv = keyv; besti = v; }  // strict > keeps first max
  }

  __shared__ float sv[256];
  __shared__ int si[256];
  sv[tid] = bestv;
  si[tid] = besti;
  __syncthreads();
  for (int s = 128; s > 0; s >>= 1) {
    if (tid < s) {
      float ov = sv[tid + s]; int oi = si[tid + s];
      if (ov > sv[tid] || (ov == sv[tid] && oi < si[tid])) {
        sv[tid] = ov; si[tid] = oi;
      }
    }
    __syncthreads();
  }

  __shared__ int tok_s;
  if (tid == 0) {
    int sampled = si[0];
    int fin = finished[b];
    int token = fin ? PAD_ : sampled;
    tokens_out[b * T_ + t] = (float)token;
    finished[b] = fin | (sampled == EOS_);
    tok_s = token;
  }
  __syncthreads();

  const int token = tok_s;
  const float* erow = embed + (size_t)token * D_;
  for (int d = tid; d < D_; d += 256) x_next[b * D_ + d] = erow[d];
}

// ---------------------------------------------------------------------------
// Host side: 1 init + 64 * (L0, L1, logits, sample) launches on `stream`.
// ---------------------------------------------------------------------------
extern "C" void kernel_launch(void* const* d_in, const int* in_sizes, int n_in,
                              void* d_out, int out_size, void* d_ws,
                              size_t ws_size, hipStream_t stream) {
  const float* input_seq = (const float*)d_in[0];
  const float* h_enc = (const float*)d_in[1];
  const float* c_enc = (const float*)d_in[2];
  const float* W_ih = (const float*)d_in[3];
  const float* W_hh = (const float*)d_in[4];
  const float* b_ih = (const float*)d_in[5];
  const float* b_hh = (const float*)d_in[6];
  const float* fc_w = (const float*)d_in[7];
  const float* fc_b = (const float*)d_in[8];
  const float* embed = (const float*)d_in[9];

  float* out = (float*)d_out;
  float* tokens_out = out + (size_t)B_ * T_ * V_;

  float* ws = (float*)d_ws;
  const int HS = B_ * H_;  // 16384 floats
  float* h0b[2] = {ws + 0 * HS, ws + 1 * HS};
  float* h1b[2] = {ws + 2 * HS, ws + 3 * HS};
  float* c0 = ws + 4 * HS;
  float* c1 = ws + 5 * HS;
  float* x = ws + 6 * HS;
  int* finished = (int*)(ws + 7 * HS);

  init_kernel<<<(HS + 255) / 256, 256, 0, stream>>>(input_seq, h_enc, c_enc,
                                                    h0b[0], h1b[0], c0, c1, x,
                                                    finished);

  const size_t WOFF = (size_t)4 * H_ * D_;  // per-layer weight stride (2048*512)
  for (int t = 0; t < T_; ++t) {
    const float* h0_in = h0b[t & 1];
    float* h0_out = h0b[(t + 1) & 1];
    const float* h1_in = h1b[t & 1];
    float* h1_out = h1b[(t + 1) & 1];

    lstm_layer_kernel<<<H_ / 16, 256, 0, stream>>>(
        x, h0_in, c0, W_ih, W_hh, b_ih, b_hh, h0_out);
    lstm_layer_kernel<<<H_ / 16, 256, 0, stream>>>(
        h0_out, h1_in, c1, W_ih + WOFF, W_hh + WOFF, b_ih + 4 * H_,
        b_hh + 4 * H_, h1_out);
    logits_kernel<<<V_ / 16, 64, 0, stream>>>(h1_out, fc_w, fc_b, out, t);
    sample_embed_kernel<<<B_, 256, 0, stream>>>(out, embed, x, finished,
                                                tokens_out, t);
  }
}